// AttentiveTransformer_60636348285210
// MI455X (gfx1250) — compile-verified
//
#include <hip/hip_runtime.h>
#include <math.h>

typedef __attribute__((ext_vector_type(2))) float v2f;
typedef __attribute__((ext_vector_type(8))) float v8f;

#define BN_EPS        1e-3f
#define ROWS_PER_BLK  32
#define D_IN          128
#define D_OUT         256
#define A_PITCH       132     // 128 + 4 pad: row stride advances bank index by 4
#define X_PITCH       260     // 256 + 4 pad: 8*260 % 64 == 32 -> halves hit disjoint banks

__device__ __forceinline__ float wave_reduce_max(float v) {
#pragma unroll
    for (int off = 16; off >= 1; off >>= 1)
        v = fmaxf(v, __shfl_xor(v, off, 32));
    return v;
}

__device__ __forceinline__ float wave_reduce_sum(float v) {
#pragma unroll
    for (int off = 16; off >= 1; off >>= 1)
        v += __shfl_xor(v, off, 32);
    return v;
}

__global__ __launch_bounds__(256)
void attentive_transformer_kernel(const float* __restrict__ inputs,
                                  const float* __restrict__ priors,
                                  const float* __restrict__ W,
                                  const float* __restrict__ gamma,
                                  const float* __restrict__ beta,
                                  const float* __restrict__ mmean,
                                  const float* __restrict__ mvar,
                                  float* __restrict__ out) {
    __shared__ float A_lds[ROWS_PER_BLK * A_PITCH];   // 16,896 B
    __shared__ float X_lds[ROWS_PER_BLK * X_PITCH];   // 33,280 B

    const int tid   = threadIdx.x;
    const int lane  = tid & 31;
    const int wave  = tid >> 5;
    const int r0    = blockIdx.x * ROWS_PER_BLK;
    const int l16   = lane & 15;
    const int lhalf = lane >> 4;

    // ---- Phase 0: stage A tile (32 x 128 f32) into LDS, fully coalesced ----
    {
        const float4* src = (const float4*)(inputs + (size_t)r0 * D_IN);
#pragma unroll
        for (int i = 0; i < 4; ++i) {
            int idx = tid + 256 * i;        // 0..1023 float4s
            int row = idx >> 5;             // 32 float4 per row
            int c4  = idx & 31;
            float4 v = src[idx];
            ((float4*)&A_lds[row * A_PITCH])[c4] = v;   // pitch*4 and c4*16 are 16B-aligned
        }
    }
    __syncthreads();

    // ---- Phase 1: WMMA f32 GEMM: each wave = 1 row-tile x 4 col-tiles ----
    const int rt = wave & 1;     // row-tile 0..1 (16 rows each)
    const int cg = wave >> 1;    // col-group 0..3 (4 tiles = 64 cols each)

    v8f acc[4] = {};

    // B-fragment base: lane holds W[k + 2*lhalf + {0,1}][cg*64 + i*16 + l16]
    const float* Wbase = W + (size_t)(2 * lhalf) * D_OUT + cg * 64 + l16;
    // A-fragment base: lane holds A[rt*16 + l16][kk + 2*lhalf + {0,1}]
    const float* Abase = &A_lds[(rt * 16 + l16) * A_PITCH + 2 * lhalf];

    for (int kk = 0; kk < D_IN; kk += 4) {
        v2f a = *(const v2f*)(Abase + kk);              // ds_load_2addr_b64, conflict-free
        const float* Wk = Wbase + (size_t)kk * D_OUT;
#pragma unroll
        for (int i = 0; i < 4; ++i) {
            v2f b;
            b.x = Wk[i * 16];
            b.y = Wk[i * 16 + D_OUT];
            acc[i] = __builtin_amdgcn_wmma_f32_16x16x4_f32(
                false, a, false, b, (short)0, acc[i], false, false);
        }
    }

    // Scatter accumulators into padded X tile (C/D layout: VGPR p -> M = p + 8*lhalf)
#pragma unroll
    for (int i = 0; i < 4; ++i) {
        int n0 = cg * 64 + i * 16;
#pragma unroll
        for (int p = 0; p < 8; ++p) {
            int row = rt * 16 + p + 8 * lhalf;
            X_lds[row * X_PITCH + n0 + l16] = acc[i][p];
        }
    }
    __syncthreads();

    // ---- Phase 2: BN + prior mask + sparsemax (one wave per 4 rows) ----
    float inv[8], shift[8];
#pragma unroll
    for (int j = 0; j < 8; ++j) {
        int c = lane + 32 * j;
        float iv = gamma[c] / sqrtf(mvar[c] + BN_EPS);
        inv[j]   = iv;
        shift[j] = beta[c] - mmean[c] * iv;
    }

    for (int rr = 0; rr < 4; ++rr) {
        int row = wave * 4 + rr;
        size_t grow = (size_t)(r0 + row) * D_OUT;

        float z[8];
        float m = -3.402823466e38f;
#pragma unroll
        for (int j = 0; j < 8; ++j) {
            int c = lane + 32 * j;
            float x = X_lds[row * X_PITCH + c];          // lanes consecutive: no conflicts
            float p = priors[grow + c];                  // coalesced 128B per j
            float zz = (x * inv[j] + shift[j]) * p;
            z[j] = zz;
            m = fmaxf(m, zz);
        }
        m = wave_reduce_max(m);

        // tau* in [m-1, m): f(tau) = sum(max(z-tau,0)) - 1 is monotone decreasing.
        // Stage 1: 10 bisection steps -> bracket width 2^-10 ~ 1e-3, so the
        // candidate support {z > lo} holds at most a few spurious members.
        float lo = m - 1.0f, hi = m;
        for (int it = 0; it < 10; ++it) {
            float mid = 0.5f * (lo + hi);
            float s = 0.0f;
#pragma unroll
            for (int j = 0; j < 8; ++j) s += fmaxf(z[j] - mid, 0.0f);
            s = wave_reduce_sum(s);
            if (s > 1.0f) lo = mid; else hi = mid;       // wave-uniform after reduce
        }

        // Stage 2: Michelot fixed-point  tau <- (sum_{z>tau} z - 1) / k.
        // Starting from lo <= tau*, tau increases monotonically and becomes
        // EXACT (the reference's closed form) once the support stabilizes;
        // 3 iterations is ample after the 1e-3 bracket.
        float tau = lo;
        for (int it = 0; it < 3; ++it) {
            float cnt = 0.0f, S = 0.0f;
#pragma unroll
            for (int j = 0; j < 8; ++j) {
                float zz = z[j];
                if (zz > tau) { cnt += 1.0f; S += zz; }
            }
            cnt = wave_reduce_sum(cnt);
            S   = wave_reduce_sum(S);
            tau = (S - 1.0f) / cnt;                      // cnt >= 1 (z_max > tau always)
        }

#pragma unroll
        for (int j = 0; j < 8; ++j)
            out[grow + lane + 32 * j] = fmaxf(z[j] - tau, 0.0f);
    }
}

extern "C" void kernel_launch(void* const* d_in, const int* in_sizes, int n_in,
                              void* d_out, int out_size, void* d_ws, size_t ws_size,
                              hipStream_t stream) {
    const float* inputs = (const float*)d_in[0];
    const float* priors = (const float*)d_in[1];
    const float* W      = (const float*)d_in[2];
    const float* gamma  = (const float*)d_in[3];
    const float* beta   = (const float*)d_in[4];
    const float* mmean  = (const float*)d_in[5];
    const float* mvar   = (const float*)d_in[6];
    float* out = (float*)d_out;

    int Btot   = in_sizes[0] / D_IN;                 // 262144 (divisible by 32)
    int blocks = Btot / ROWS_PER_BLK;

    hipLaunchKernelGGL(attentive_transformer_kernel, dim3(blocks), dim3(256), 0, stream,
                       inputs, priors, W, gamma, beta, mmean, mvar, out);
}